// LocalAggregator_64587718197537
// MI455X (gfx1250) — compile-verified
//
#include <hip/hip_runtime.h>
#include <hip/hip_bf16.h>

// ---------------------------------------------------------------------------
// Problem constants (match reference)
// ---------------------------------------------------------------------------
#define H_ 60
#define W_ 60
#define D_ 36
#define NPTS (H_ * W_ * D_)      // 129600 query points
#define P_ 2048                  // Gaussians
#define PPAD 2056                // table entries incl. sentinel + padding
#define C_ 13                    // semantic channels
#define GRID_ 0.08f
#define SCALE_MULT_ 3.0f
#define LOG2E_HALF 0.72134752044448170f   // 0.5 * log2(e)

typedef __attribute__((ext_vector_type(2))) float v2f;
typedef __attribute__((ext_vector_type(8))) float v8f;

// Gaussian record: 12 floats (48 B) = 3 x float4, loaded as one clause.
//   [0..3]  mx, my, mz, opacity
//   [4..7]  ixx, iyy, izz, ixy
//   [8..11] iyz, ixz, bitcast(pack), 0        pack = r<<24 | mz<<16 | my<<8 | mx
// (m_int in [1,58], radius in [2,8] for this problem -> u8 fields are exact)

// ---------------------------------------------------------------------------
// Prep kernel: build interleaved records + u32 cull table + padded semantics.
// One thread per Gaussian; thread P_ writes a sentinel whose record yields an
// exact-zero weight (m=255 -> inbox false for all in-range points; opacity 0).
// ---------------------------------------------------------------------------
__global__ __launch_bounds__(256) void la_prep_kernel(
    const float* __restrict__ means,    // [P,3]
    const float* __restrict__ opac,     // [P]
    const float* __restrict__ sem,      // [P,13]
    const float* __restrict__ scales,   // [P,3]
    const float* __restrict__ cov,      // [P,3,3]
    const float* __restrict__ origin,   // [3]
    float* __restrict__ ggeo,           // [PPAD*12]
    unsigned* __restrict__ gcull,       // [PPAD]
    float* __restrict__ gsem)           // [PPAD,16]
{
    int p = blockIdx.x * blockDim.x + threadIdx.x;
    if (p > P_) return;

    if (p == P_) {                      // sentinel record
        const unsigned pack = (0u << 24) | (255u << 16) | (255u << 8) | 255u;
#pragma unroll
        for (int i = 0; i < 12; ++i) ggeo[p * 12 + i] = 0.0f;
        ggeo[p * 12 + 10] = __uint_as_float(pack);
        gcull[p] = pack;
#pragma unroll
        for (int cc = 0; cc < 16; ++cc) gsem[p * 16 + cc] = 0.0f;
        return;
    }

    // packed = (xx, yy, zz, xy, yz, xz) = cov[0,4,8,1,5,2]
    float a = cov[p * 9 + 0], b = cov[p * 9 + 4], c = cov[p * 9 + 8];
    float d = cov[p * 9 + 1], e = cov[p * 9 + 5], f = cov[p * 9 + 2];
    float det = a * (b * c - e * e) - d * (d * c - e * f) + f * (d * e - b * f);
    float ixx = (b * c - e * e) / det;
    float iyy = (a * c - f * f) / det;
    float izz = (a * b - d * d) / det;
    float ixy = (e * f - d * c) / det;
    float iyz = (d * f - a * e) / det;
    float ixz = (d * e - b * f) / det;

    float mx = means[p * 3 + 0], my = means[p * 3 + 1], mz = means[p * 3 + 2];
    float s0 = scales[p * 3 + 0], s1 = scales[p * 3 + 1], s2 = scales[p * 3 + 2];
    float smax = fmaxf(s0, fmaxf(s1, s2));
    int radius = (int)ceilf(smax * SCALE_MULT_ / GRID_);

    float ox = origin[0], oy = origin[1], oz = origin[2];
    int gx = (int)((mx - ox) / GRID_);   // trunc toward zero, matches astype(int32)
    int gy = (int)((my - oy) / GRID_);
    int gz = (int)((mz - oz) / GRID_);

    const unsigned pack = ((unsigned)radius << 24) | ((unsigned)(gz & 0xff) << 16)
                        | ((unsigned)(gy & 0xff) << 8) | (unsigned)(gx & 0xff);

    float* rec = ggeo + p * 12;
    rec[0] = mx;  rec[1] = my;  rec[2] = mz;  rec[3] = opac[p];
    rec[4] = ixx; rec[5] = iyy; rec[6] = izz; rec[7] = ixy;
    rec[8] = iyz; rec[9] = ixz; rec[10] = __uint_as_float(pack); rec[11] = 0.0f;
    gcull[p] = pack;

#pragma unroll
    for (int cc = 0; cc < 16; ++cc)
        gsem[p * 16 + cc] = (cc < C_) ? sem[p * C_ + cc] : 0.0f;
}

// ---------------------------------------------------------------------------
// Main kernel: each wave owns a 16-point x 16-channel tile.
// Phase 1: cull 2048 Gaussians against the tile's integer-bin bbox (exact-
//          zero pairs only), compacting survivor indices into a per-wave LDS
//          list via wave32 ballot + prefix popcount; pad with 8 sentinels so
//          the compute loop is branch-free.
// Phase 2: D = W(16 x nsurv) x S(nsurv x 16) via V_WMMA_F32_16X16X4_F32,
//          2 groups (8 survivors, 2 WMMAs) per iteration. One ds_load_b32
//          yields a lane's two slot indices; one 48-B record = 3 clause-
//          grouped b128 loads (half-wave broadcast, L2/WGP$-resident).
// Block = 128 threads = 4 independent waves = 64 points.
// ---------------------------------------------------------------------------
__global__ __launch_bounds__(128) void la_agg_kernel(
    const float* __restrict__ pts,     // [N,3]
    const float* __restrict__ origin,  // [3]
    const float* __restrict__ ggeo,    // [PPAD*12]
    const unsigned* __restrict__ gcull,// [PPAD]
    const float* __restrict__ gsem,    // [PPAD,16]
    float* __restrict__ out)           // [N,13]
{
    __shared__ unsigned short s_list[4][P_ + 8];   // per-wave survivor lists

    const int tid  = threadIdx.x;
    const int lane = tid & 31;
    const int wave = tid >> 5;
    const int n    = lane & 15;        // channel index (B/D) == point-in-tile (A)
    const int half = lane >> 4;        // 0: K=0,1   1: K=2,3  (A & B layouts)
    const int koff = half << 1;

    const int tile  = blockIdx.x * 4 + wave;
    const int pbase = tile * 16;
    const int mypt  = pbase + n;       // lanes L and L+16 share the same point

    const float ox = origin[0], oy = origin[1], oz = origin[2];
    const float px = pts[mypt * 3 + 0];
    const float py = pts[mypt * 3 + 1];
    const float pz = pts[mypt * 3 + 2];
    const int pix = (int)((px - ox) / GRID_);
    const int piy = (int)((py - oy) / GRID_);
    const int piz = (int)((pz - oz) / GRID_);

    // ---- tile bbox of integer bins; lane halves hold identical points, so a
    //      16-lane xor reduction (masks 8,4,2,1) is wave-complete.
    int bxmin = pix, bxmax = pix;
    int bymin = piy, bymax = piy;
    int bzmin = piz, bzmax = piz;
#pragma unroll
    for (int m = 8; m >= 1; m >>= 1) {
        bxmin = min(bxmin, __shfl_xor(bxmin, m, 32));
        bxmax = max(bxmax, __shfl_xor(bxmax, m, 32));
        bymin = min(bymin, __shfl_xor(bymin, m, 32));
        bymax = max(bymax, __shfl_xor(bymax, m, 32));
        bzmin = min(bzmin, __shfl_xor(bzmin, m, 32));
        bzmax = max(bzmax, __shfl_xor(bzmax, m, 32));
    }

    // ---- Phase 1: cull + compact (32 Gaussians per step, coalesced b32 loads)
    int nsurv = 0;
    for (int base = 0; base < P_; base += 32) {
        const int g = base + lane;
        const unsigned cp = gcull[g];
        const int gx = (int)(cp & 0xffu);
        const int gy = (int)((cp >> 8) & 0xffu);
        const int gz = (int)((cp >> 16) & 0xffu);
        const int r  = (int)(cp >> 24);
        const bool hit = (gx - r <= bxmax) & (gx + r >= bxmin)
                       & (gy - r <= bymax) & (gy + r >= bymin)
                       & (gz - r <= bzmax) & (gz + r >= bzmin);
        const unsigned mask = __builtin_amdgcn_ballot_w32(hit);
        if (hit) {
            const int ofs = __popc(mask & ((1u << lane) - 1u));
            s_list[wave][nsurv + ofs] = (unsigned short)g;
        }
        nsurv += __popc(mask);
    }
    // pad with 8 sentinel entries -> compute loop needs no tail predicate
    if (lane < 8) s_list[wave][nsurv + lane] = (unsigned short)P_;

    // ---- Phase 2: accumulate over survivors, 8 per iteration (2 WMMAs)
    v8f acc = {};   // 16x16 f32 accumulator tile
    const int niter = (nsurv + 7) >> 3;
    const unsigned short* mylist = &s_list[wave][0];

    for (int j = 0; j < niter; ++j) {
#pragma unroll
        for (int u = 0; u < 2; ++u) {
            const int sbase = (j << 3) + (u << 2) + koff;   // 4B-aligned u16 pair
            const unsigned packidx = *(const unsigned*)(mylist + sbase);
            v2f aw, bv;
#pragma unroll
            for (int t = 0; t < 2; ++t) {
                const int g = (t == 0) ? (int)(packidx & 0xffffu)
                                       : (int)(packidx >> 16);
                const float4* rec = (const float4*)(ggeo + g * 12);
                const float4 r0 = rec[0];   // mx, my, mz, opacity
                const float4 r1 = rec[1];   // ixx, iyy, izz, ixy
                const float4 r2 = rec[2];   // iyz, ixz, pack, 0
                const float dx = px - r0.x, dy = py - r0.y, dz = pz - r0.z;
                const float quad = r1.x * dx * dx + r1.y * dy * dy + r1.z * dz * dz
                                 + 2.0f * (r1.w * dx * dy + r2.x * dy * dz + r2.y * dx * dz);
                const unsigned cp = __float_as_uint(r2.z);
                const int gx = (int)(cp & 0xffu);
                const int gy = (int)((cp >> 8) & 0xffu);
                const int gz = (int)((cp >> 16) & 0xffu);
                const int r  = (int)(cp >> 24);
                int ax = pix - gx; ax = ax < 0 ? -ax : ax;
                int ay = piy - gy; ay = ay < 0 ? -ay : ay;
                int az = piz - gz; az = az < 0 ? -az : az;
                const bool inbox = (ax <= r) & (ay <= r) & (az <= r);
                const float w = r0.w * __builtin_amdgcn_exp2f(-LOG2E_HALF * quad);
                aw[t] = inbox ? w : 0.0f;   // exact 0 outside box; sentinel -> 0
                bv[t] = gsem[g * 16 + n];   // B row K, col N = n
            }
            // D(16x16,f32) += A(16x4,f32) x B(4x16,f32)
            acc = __builtin_amdgcn_wmma_f32_16x16x4_f32(
                /*neg_a=*/false, aw, /*neg_b=*/false, bv,
                /*c_mod=*/(short)0, acc, /*reuse_a=*/false, /*reuse_b=*/false);
        }
    }

    // C/D layout: VGPR r -> M = r + 8*half, N = lane&15
#pragma unroll
    for (int r = 0; r < 8; ++r) {
        const int point = pbase + r + half * 8;
        if (n < C_) out[point * C_ + n] = acc[r];
    }
}

// ---------------------------------------------------------------------------
// Launch
// ---------------------------------------------------------------------------
extern "C" void kernel_launch(void* const* d_in, const int* in_sizes, int n_in,
                              void* d_out, int out_size, void* d_ws, size_t ws_size,
                              hipStream_t stream) {
    const float* pts    = (const float*)d_in[0];  // [1,N,3]
    const float* means  = (const float*)d_in[1];  // [1,P,3]
    const float* opac   = (const float*)d_in[2];  // [1,P]
    const float* sem    = (const float*)d_in[3];  // [1,P,13]
    const float* scales = (const float*)d_in[4];  // [1,P,3]
    const float* cov    = (const float*)d_in[5];  // [1,P,3,3]
    const float* origin = (const float*)d_in[6];  // [3]
    float* out = (float*)d_out;                   // [N,13]

    // Workspace carve-out (16B-aligned offsets)
    char* ws = (char*)d_ws;
    float*    ggeo  = (float*   )(ws + 0);        // PPAD*48 = 98688 B
    unsigned* gcull = (unsigned*)(ws + 98688);    // PPAD*4  =  8224 B
    float*    gsem  = (float*   )(ws + 106912);   // PPAD*64 = 131584 B
    // total = 238496 B

    la_prep_kernel<<<(P_ + 256) / 256, 256, 0, stream>>>(
        means, opac, sem, scales, cov, origin, ggeo, gcull, gsem);

    // 129600 points / (4 waves * 16 points) = 2025 blocks exactly
    la_agg_kernel<<<NPTS / 64, 128, 0, stream>>>(
        pts, origin, ggeo, gcull, gsem, out);
}